// ElmanRNN_pred_v3_10728828305917
// MI455X (gfx1250) — compile-verified
//
#include <hip/hip_runtime.h>
#include <hip/hip_bf16.h>

// ---- problem constants (from reference) ----
#define B_    64
#define T_    2048
#define DIN_  64
#define H_    128
#define DOUT_ 64
#define KREC  2
#define TS_   (T_ - KREC)   // 2046 scan steps

typedef float v2f __attribute__((ext_vector_type(2)));
typedef float v8f __attribute__((ext_vector_type(8)));

__device__ __forceinline__ v8f wmma4(v2f a, v2f b, v8f c) {
    // D = A(16x4) * B(4x16) + C, fp32 — V_WMMA_F32_16X16X4_F32
    return __builtin_amdgcn_wmma_f32_16x16x4_f32(false, a, false, b, (short)0, c, false, false);
}

__device__ __forceinline__ float fast_tanh(float x) {
    // tanh(x) = 1 - 2/(exp(2x)+1); saturates correctly for |x| large
    float e = __expf(2.0f * x);
    return 1.0f - 2.0f * __builtin_amdgcn_rcpf(e + 1.0f);
}

// =====================================================================
// Kernel 1: serial Elman scan, single workgroup (8 waves). Wh/Wi WMMA
// B-fragments live in registers for the whole scan; hidden state is
// double-buffered in LDS; next step's x is prefetched into WGP cache.
// Writes z[:, t+K, :] and hfin.
// =====================================================================
extern "C" __global__ void __launch_bounds__(256)
rnn_scan_kernel(const float* __restrict__ x,  const float* __restrict__ h0,
                const float* __restrict__ Wi, const float* __restrict__ bi,
                const float* __restrict__ Wh, const float* __restrict__ bh,
                float* __restrict__ z, float* __restrict__ hfin)
{
    extern __shared__ float smem[];
    float* sWh = smem;                     // H*H      = 16384 f (staging only)
    float* sWi = sWh + H_ * H_;            // H*DIN    =  8192 f (staging only)
    float* sHa = sWi + H_ * DIN_;          // B*H      =  8192 f
    float* sHb = sHa + B_ * H_;            // B*H      =  8192 f
    float* sB1 = sHb + B_ * H_;            // H  (bi+bh)
    float* sB2 = sB1 + H_;                 // H  (bh)

    const int tid = threadIdx.x;
    for (int i = tid; i < H_ * H_;   i += 256) sWh[i] = Wh[i];
    for (int i = tid; i < H_ * DIN_; i += 256) sWi[i] = Wi[i];
    for (int i = tid; i < B_ * H_;   i += 256) sHa[i] = h0[i];
    if (tid < H_) { sB1[tid] = bi[tid] + bh[tid]; sB2[tid] = bh[tid]; }
    __syncthreads();

    const int w   = tid >> 5;        // wave 0..7 -> N-slice
    const int l   = tid & 31;
    const int lo  = l & 15;
    const int hi  = l >> 4;          // 0/1 half of wave
    const int kb2 = 2 * hi;          // k offset inside a K=4 slab
    const int n   = w * 16 + lo;     // this lane's output column

    const float bias1 = sB1[n];
    const float bias2 = sB2[n];

    // ---- hoist loop-invariant WMMA B-fragments into registers ----
    v2f bWh[H_ / 4];                 // 32 x v2f = 64 VGPR
    #pragma unroll
    for (int ks = 0; ks < H_ / 4; ++ks)
        bWh[ks] = *(const v2f*)(sWh + n * H_ + ks * 4 + kb2);
    v2f bWi[DIN_ / 4];               // 16 x v2f = 32 VGPR
    #pragma unroll
    for (int ks = 0; ks < DIN_ / 4; ++ks)
        bWi[ks] = *(const v2f*)(sWi + n * DIN_ + ks * 4 + kb2);

    float* cur = sHa;
    float* nxt = sHb;

    // acc += hsrc(64xH) * Wh^T, this wave's N-slice, 4 M-tiles
    auto wh_mm = [&](const float* hsrc, v8f& c0, v8f& c1, v8f& c2, v8f& c3) {
        #pragma unroll
        for (int ks = 0; ks < H_ / 4; ++ks) {
            const int kb = ks * 4 + kb2;
            v2f a0 = *(const v2f*)(hsrc + (lo     ) * H_ + kb);
            v2f a1 = *(const v2f*)(hsrc + (16 + lo) * H_ + kb);
            v2f a2 = *(const v2f*)(hsrc + (32 + lo) * H_ + kb);
            v2f a3 = *(const v2f*)(hsrc + (48 + lo) * H_ + kb);
            c0 = wmma4(a0, bWh[ks], c0);
            c1 = wmma4(a1, bWh[ks], c1);
            c2 = wmma4(a2, bWh[ks], c2);
            c3 = wmma4(a3, bWh[ks], c3);
        }
    };

    auto store_tanh = [&](v8f& acc, int mt, float bias, float* dst) {
        #pragma unroll
        for (int r = 0; r < 8; ++r) {
            int m = mt * 16 + r + 8 * hi;
            dst[m * H_ + n] = fast_tanh(acc[r] + bias);
        }
    };

    for (int t = 0; t < TS_; ++t) {
        // ---------- phase 1: h = tanh(x_t Wi^T + h Wh^T + (bi+bh)) ----------
        v8f c0 = {}, c1 = {}, c2 = {}, c3 = {};
        #pragma unroll
        for (int ks = 0; ks < DIN_ / 4; ++ks) {
            const int kb = ks * 4 + kb2;
            v2f a0 = *(const v2f*)(x + ((size_t)(lo     ) * T_ + t) * DIN_ + kb);
            v2f a1 = *(const v2f*)(x + ((size_t)(16 + lo) * T_ + t) * DIN_ + kb);
            v2f a2 = *(const v2f*)(x + ((size_t)(32 + lo) * T_ + t) * DIN_ + kb);
            v2f a3 = *(const v2f*)(x + ((size_t)(48 + lo) * T_ + t) * DIN_ + kb);
            c0 = wmma4(a0, bWi[ks], c0);
            c1 = wmma4(a1, bWi[ks], c1);
            c2 = wmma4(a2, bWi[ks], c2);
            c3 = wmma4(a3, bWi[ks], c3);
        }
        // prefetch next step's x rows into WGP cache (global_prefetch_b8)
        if (t + 1 < TS_) {
            const float* p0 = x + ((size_t)(lo     ) * T_ + (t + 1)) * DIN_;
            const float* p1 = x + ((size_t)(16 + lo) * T_ + (t + 1)) * DIN_;
            const float* p2 = x + ((size_t)(32 + lo) * T_ + (t + 1)) * DIN_;
            const float* p3 = x + ((size_t)(48 + lo) * T_ + (t + 1)) * DIN_;
            __builtin_prefetch(p0, 0, 1);  __builtin_prefetch(p0 + 32, 0, 1);
            __builtin_prefetch(p1, 0, 1);  __builtin_prefetch(p1 + 32, 0, 1);
            __builtin_prefetch(p2, 0, 1);  __builtin_prefetch(p2 + 32, 0, 1);
            __builtin_prefetch(p3, 0, 1);  __builtin_prefetch(p3 + 32, 0, 1);
        }
        wh_mm(cur, c0, c1, c2, c3);
        store_tanh(c0, 0, bias1, nxt);
        store_tanh(c1, 1, bias1, nxt);
        store_tanh(c2, 2, bias1, nxt);
        store_tanh(c3, 3, bias1, nxt);
        __syncthreads();
        { float* tmp = cur; cur = nxt; nxt = tmp; }

        // ---------- phase 2: first extra recurrence ----------
        c0 = (v8f){}; c1 = (v8f){}; c2 = (v8f){}; c3 = (v8f){};
        wh_mm(cur, c0, c1, c2, c3);
        store_tanh(c0, 0, bias2, nxt);
        store_tanh(c1, 1, bias2, nxt);
        store_tanh(c2, 2, bias2, nxt);
        store_tanh(c3, 3, bias2, nxt);
        __syncthreads();
        { float* tmp = cur; cur = nxt; nxt = tmp; }

        // ---------- phase 3: second extra recurrence + emit z[:, t+K, :] ----
        c0 = (v8f){}; c1 = (v8f){}; c2 = (v8f){}; c3 = (v8f){};
        wh_mm(cur, c0, c1, c2, c3);
        {
            float* zslice = z + (size_t)(t + KREC) * H_;       // + m*T*H below
            float* hf = (t == TS_ - 1) ? hfin : nullptr;
            #pragma unroll
            for (int mt = 0; mt < 4; ++mt) {
                v8f& acc = (mt == 0) ? c0 : (mt == 1) ? c1 : (mt == 2) ? c2 : c3;
                #pragma unroll
                for (int r = 0; r < 8; ++r) {
                    int m = mt * 16 + r + 8 * hi;
                    float hv = fast_tanh(acc[r] + bias2);
                    nxt[m * H_ + n] = hv;
                    zslice[(size_t)m * (T_ * H_) + n] = hv;
                    if (hf) hf[m * H_ + n] = hv;
                }
            }
        }
        __syncthreads();
        { float* tmp = cur; cur = nxt; nxt = tmp; }
    }
}

// =====================================================================
// Kernel 2: out = softmax(z @ W3^T + b3) over last dim. 64 rows / block.
// Rows with t < KREC use z == 0 (never reads poisoned workspace there).
// =====================================================================
extern "C" __global__ void __launch_bounds__(128)
proj_softmax_kernel(const float* __restrict__ z, const float* __restrict__ W3,
                    const float* __restrict__ b3, float* __restrict__ out)
{
    __shared__ float sW3[DOUT_ * H_];     // 32 KB
    __shared__ float sB3[DOUT_];
    __shared__ float sLog[64 * 65];       // padded stride 65 -> no bank conflicts
    __shared__ float sMax[64];
    __shared__ float sInv[64];

    const int tid = threadIdx.x;
    for (int i = tid; i < DOUT_ * H_; i += 128) sW3[i] = W3[i];
    if (tid < DOUT_) sB3[tid] = b3[tid];
    __syncthreads();

    const int w   = tid >> 5;           // wave 0..3 -> 16-row slice
    const int l   = tid & 31;
    const int lo  = l & 15;
    const int hi  = l >> 4;
    const int kb2 = 2 * hi;
    const int rowBlock = blockIdx.x * 64;
    const int m    = rowBlock + w * 16 + lo;   // global row = b*T + t
    const int tcol = m & (T_ - 1);             // T_ is a power of two

    v8f c0 = {}, c1 = {}, c2 = {}, c3 = {};
    #pragma unroll 8
    for (int ks = 0; ks < H_ / 4; ++ks) {
        const int kb = ks * 4 + kb2;
        v2f a;
        if (tcol >= KREC) a = *(const v2f*)(z + (size_t)m * H_ + kb);
        else { a[0] = 0.0f; a[1] = 0.0f; }
        v2f b0 = *(const v2f*)(sW3 + (lo     ) * H_ + kb);
        v2f b1 = *(const v2f*)(sW3 + (16 + lo) * H_ + kb);
        v2f b2 = *(const v2f*)(sW3 + (32 + lo) * H_ + kb);
        v2f b3f = *(const v2f*)(sW3 + (48 + lo) * H_ + kb);
        c0 = wmma4(a, b0, c0);
        c1 = wmma4(a, b1, c1);
        c2 = wmma4(a, b2, c2);
        c3 = wmma4(a, b3f, c3);
    }

    // spill logits (+bias) to LDS
    #pragma unroll
    for (int r = 0; r < 8; ++r) {
        const int rl = w * 16 + r + 8 * hi;
        sLog[rl * 65 +      lo] = c0[r] + sB3[     lo];
        sLog[rl * 65 + 16 + lo] = c1[r] + sB3[16 + lo];
        sLog[rl * 65 + 32 + lo] = c2[r] + sB3[32 + lo];
        sLog[rl * 65 + 48 + lo] = c3[r] + sB3[48 + lo];
    }
    __syncthreads();

    // per-row max and 1/sum(exp)
    if (tid < 64) {
        float mx = -3.4e38f;
        #pragma unroll 8
        for (int o = 0; o < DOUT_; ++o) mx = fmaxf(mx, sLog[tid * 65 + o]);
        float s = 0.0f;
        #pragma unroll 8
        for (int o = 0; o < DOUT_; ++o) s += __expf(sLog[tid * 65 + o] - mx);
        sMax[tid] = mx;
        sInv[tid] = 1.0f / s;
    }
    __syncthreads();

    // coalesced normalized writes
    for (int idx = tid; idx < 64 * DOUT_; idx += 128) {
        const int rl = idx >> 6;
        const int o  = idx & 63;
        out[(size_t)(rowBlock + rl) * DOUT_ + o] =
            __expf(sLog[rl * 65 + o] - sMax[rl]) * sInv[rl];
    }
}

extern "C" void kernel_launch(void* const* d_in, const int* in_sizes, int n_in,
                              void* d_out, int out_size, void* d_ws, size_t ws_size,
                              hipStream_t stream) {
    (void)in_sizes; (void)n_in; (void)out_size; (void)ws_size;
    const float* x  = (const float*)d_in[0];
    const float* h0 = (const float*)d_in[1];
    const float* Wi = (const float*)d_in[2];
    const float* bi = (const float*)d_in[3];
    const float* Wh = (const float*)d_in[4];
    const float* bh = (const float*)d_in[5];
    const float* W3 = (const float*)d_in[6];
    const float* b3 = (const float*)d_in[7];

    float* out  = (float*)d_out;                       // (B,T,DOUT)
    float* hfin = out + (size_t)B_ * T_ * DOUT_;       // (B,H) tail of d_out
    float* z    = (float*)d_ws;                        // (B,T,H) fp32 scratch

    const size_t lds1 =
        (size_t)(H_ * H_ + H_ * DIN_ + 2 * B_ * H_ + 2 * H_) * sizeof(float);

    rnn_scan_kernel<<<dim3(1), dim3(256), lds1, stream>>>(
        x, h0, Wi, bi, Wh, bh, z, hfin);

    proj_softmax_kernel<<<dim3((B_ * T_) / 64), dim3(128), 0, stream>>>(
        z, W3, b3, out);
}